// SinabsLIFModel_30064771072448
// MI455X (gfx1250) — compile-verified
//
#include <hip/hip_runtime.h>
#include <cstdint>

// f32 WMMA fragment types: A/B = 16x4 / 4x16 f32 (2 VGPRs), C/D = 16x16 f32 (8 VGPRs)
typedef __attribute__((ext_vector_type(2))) float v2f;
typedef __attribute__((ext_vector_type(8))) float v8f;

#define LIF_ALPHA 0.95122942450071400910f   // exp(-1/20)
#define LIF_ONEM  0.04877057549928599090f   // 1 - alpha

// ---------------------------------------------------------------------------
// GEMM: Y[M, NOUT] = X[M, K] @ W^T, W is [NOUT, K] row-major.
// One wave per 16-row m-tile. Chained V_WMMA_F32_16X16X4_F32 (exact fp32).
// ---------------------------------------------------------------------------
template<int K, int NOUT>
__global__ __launch_bounds__(256) void gemm_wmma_kernel(
    const float* __restrict__ X, const float* __restrict__ W,
    float* __restrict__ Y, int M)
{
  constexpr int NT = (NOUT + 15) / 16;   // N tiles of 16
  constexpr int KS = K / 4;              // K slices of 4
  const int lane = threadIdx.x & 31;
  const int wave = (blockIdx.x * blockDim.x + threadIdx.x) >> 5;
  const int m0   = wave * 16;
  if (m0 >= M) return;                   // wave-uniform, EXEC stays all-ones
  const int half = lane >> 4;
  const int l    = lane & 15;

  // A fragments: 16x4 f32 layout -> lane = M, VGPR0={K0|K2}, VGPR1={K1|K3}
  v2f a[KS];
  const float* xrow = X + (size_t)(m0 + l) * K;
  #pragma unroll
  for (int s = 0; s < KS; ++s)
    a[s] = *(const v2f*)(xrow + 4 * s + 2 * half);

  // B fragments: B[k][n] = W[n][k]; lane = N, mirrored half-wave K pairing
  v2f b[NT][KS];
  #pragma unroll
  for (int nt = 0; nt < NT; ++nt) {
    const int n = nt * 16 + l;
    #pragma unroll
    for (int s = 0; s < KS; ++s) {
      v2f bv = {0.0f, 0.0f};
      if (n < NOUT) bv = *(const v2f*)(W + (size_t)n * K + 4 * s + 2 * half);
      b[nt][s] = bv;                     // padded rows contribute exact zeros
    }
  }

  #pragma unroll
  for (int nt = 0; nt < NT; ++nt) {
    v8f acc = {};
    #pragma unroll
    for (int s = 0; s < KS; ++s)
      acc = __builtin_amdgcn_wmma_f32_16x16x4_f32(
          false, a[s], false, b[nt][s], (short)0, acc, false, false);

    const int n = nt * 16 + l;
    if (n < NOUT) {
      #pragma unroll
      for (int r = 0; r < 8; ++r)
        Y[(size_t)(m0 + 8 * half + r) * NOUT + n] = acc[r];
    }
  }
}

// ---------------------------------------------------------------------------
// LIF scan with CDNA5 async double-buffered LDS staging.
//   v[t] = a*v + (1-a)*x[t]; s = max(floor(v),0); v -= s   (theta = 1)
// Workgroup owns WB batches x N neurons. Time is processed in 32-step chunks:
//   - global_load_async_to_lds_b128 (ASYNCcnt) fills the next LDS buffer
//   - global_prefetch warms L2 eight chunks ahead (no counters, deep pipeline)
//   - scan reads current buffer via ds_load, spikes stored straight to HBM
// ---------------------------------------------------------------------------
template<int N, int WB, int NTHREADS>
__global__ __launch_bounds__(NTHREADS) void lif_scan_async_kernel(
    const float* __restrict__ X, float* __restrict__ S, int T)
{
  constexpr int TCH = 32;                              // timesteps per chunk
  constexpr int CHUNK_FLOATS = WB * TCH * N;           // floats per buffer
  constexpr int UNITS = CHUNK_FLOATS / 4;              // 16B units per chunk
  constexpr int UPT   = UNITS / NTHREADS;              // async loads per thread (8)
  constexpr int UPB   = TCH * N / 4;                   // 16B units per batch slab
  constexpr int LINES_PER_SLAB = TCH * N / 32;         // 128B lines per batch slab
  constexpr int PF = 8;                                // prefetch depth (chunks)

  __shared__ float lds[2 * CHUNK_FLOATS];              // <= 64KB static

  const int tid = threadIdx.x;
  const int b0  = blockIdx.x * WB;
  const int bl  = tid / N;
  const int n   = tid - bl * N;
  const int NC  = T / TCH;

  // fire UPT async 16B loads for chunk c into LDS buffer `buf`
  auto issue = [&](int c, int buf) {
    const uint32_t lds_base =
        (uint32_t)(uintptr_t)lds + (uint32_t)(buf * CHUNK_FLOATS * 4);
    #pragma unroll
    for (int k = 0; k < UPT; ++k) {
      const int u  = tid + k * NTHREADS;
      const int ub = u / UPB;                          // batch slab
      const int uo = u - ub * UPB;                     // unit within slab
      const float* g =
          X + ((size_t)(b0 + ub) * T + (size_t)c * TCH) * N + (size_t)uo * 4;
      const uint32_t l = lds_base + (uint32_t)(u * 16);
      asm volatile("global_load_async_to_lds_b128 %0, %1, off"
                   :: "v"(l), "v"(g) : "memory");
    }
  };

  float v = 0.0f;
  issue(0, 0);
  for (int c = 0; c < NC; ++c) {
    const int cur = c & 1;
    if (c + 1 < NC) issue(c + 1, 1 - cur);             // fill other buffer

    if (c + PF < NC) {                                 // warm L2 far ahead
      const int slab = tid / LINES_PER_SLAB;
      const int line = tid - slab * LINES_PER_SLAB;
      __builtin_prefetch(
          X + ((size_t)(b0 + slab) * T + (size_t)(c + PF) * TCH) * N + line * 32,
          0, 0);
    }

    // retire chunk c's loads (async loads complete in order per wave)
    if (c + 1 < NC) asm volatile("s_wait_asynccnt %0" :: "i"(UPT) : "memory");
    else            asm volatile("s_wait_asynccnt 0" ::: "memory");
    __syncthreads();

    const float* buf = lds + cur * CHUNK_FLOATS + (size_t)(bl * TCH) * N + n;
    float* op = S + ((size_t)(b0 + bl) * T + (size_t)c * TCH) * N + n;
    #pragma unroll 4
    for (int tl = 0; tl < TCH; ++tl) {
      const float x = buf[(size_t)tl * N];             // ds_load_b32
      v = LIF_ALPHA * v + LIF_ONEM * x;
      const float s = fmaxf(floorf(v), 0.0f);          // MultiSpike, theta = 1
      v -= s;                                          // membrane-subtract reset
      op[(size_t)tl * N] = s;
    }
    __syncthreads();                                   // buffer reused next iter
  }
}

// ---------------------------------------------------------------------------
extern "C" void kernel_launch(void* const* d_in, const int* in_sizes, int n_in,
                              void* d_out, int out_size, void* d_ws, size_t ws_size,
                              hipStream_t stream) {
  (void)in_sizes; (void)n_in; (void)out_size; (void)ws_size;
  const float* data = (const float*)d_in[0];   // [256, 4096, 16]
  const float* w1   = (const float*)d_in[1];   // [16, 16]
  const float* w2   = (const float*)d_in[2];   // [32, 16]
  const float* w3   = (const float*)d_in[3];   // [10, 32]
  float* out = (float*)d_out;                  // [256, 4096, 10]

  const int B = 256, T = 4096;
  const int M = B * T;                         // 1,048,576 rows

  // Two M*32-float regions; dead-buffer aliasing keeps scratch at 256 MB.
  float* regA = (float*)d_ws;
  float* regB = regA + (size_t)M * 32;
  float* y1 = regA;                     // [M,16] gemm1 out
  float* s1 = regA + (size_t)M * 16;    // [M,16] spikes1
  float* y2 = regB;                     // [M,32] gemm2 out
  float* s2 = regA;                     // [M,32] spikes2 (y1,s1 dead)
  float* y3 = regB;                     // [M,10] gemm3 out (y2 dead)

  const dim3 gemm_grid((M / 16) * 32 / 256);   // one wave per 16-row m-tile

  gemm_wmma_kernel<16, 16><<<gemm_grid, 256, 0, stream>>>(data, w1, y1, M);
  lif_scan_async_kernel<16, 16, 256><<<B / 16, 256, 0, stream>>>(y1, s1, T);
  gemm_wmma_kernel<16, 32><<<gemm_grid, 256, 0, stream>>>(s1, w2, y2, M);
  lif_scan_async_kernel<32, 8, 256><<<B / 8, 256, 0, stream>>>(y2, s2, T);
  gemm_wmma_kernel<32, 10><<<gemm_grid, 256, 0, stream>>>(s2, w3, y3, M);
  lif_scan_async_kernel<10, 16, 160><<<B / 16, 160, 0, stream>>>(y3, out, T);
}